// NerfactoRenderer_55465207661109
// MI455X (gfx1250) — compile-verified
//
#include <hip/hip_runtime.h>
#include <stdint.h>

// ---------------------------------------------------------------------------
// NeRF volume rendering weights (render_weights + weighted sums) for MI455X.
// Memory-bound streaming kernel: stage inputs into LDS with CDNA5 async
// global->LDS loads (ASYNCcnt), per-thread sequential scan per ray, drain
// weights with async LDS->global stores.
// ---------------------------------------------------------------------------

#define S_SAMPLES 48
#define RAYS      64            // rays per block == threads per block
#define CHUNK     16            // samples per chunk
#define NCHUNK    3             // 48 / 16

// LDS byte strides per ray (padded: 16B-aligned, low bank conflicts)
#define POS_STRIDE 208          // 52 floats (48 chunk + 3 boundary + pad)
#define RGB_STRIDE 208          // 52 floats
#define DEN_STRIDE 80           // 20 floats (16 + pad)
#define W_STRIDE   80           // 20 floats

#define LDS_POS 0
#define LDS_RGB (LDS_POS + RAYS * POS_STRIDE)   // 13312
#define LDS_DEN (LDS_RGB + RAYS * RGB_STRIDE)   // 26624
#define LDS_W   (LDS_DEN + RAYS * DEN_STRIDE)   // 31744
#define LDS_TOTAL (LDS_W + RAYS * W_STRIDE)     // 36864 bytes

__device__ __forceinline__ void async_load_b128(uint32_t lds_off, const void* gaddr) {
    asm volatile("global_load_async_to_lds_b128 %0, %1, off"
                 :: "v"(lds_off), "v"(gaddr) : "memory");
}
__device__ __forceinline__ void async_load_b32(uint32_t lds_off, const void* gaddr) {
    asm volatile("global_load_async_to_lds_b32 %0, %1, off"
                 :: "v"(lds_off), "v"(gaddr) : "memory");
}
__device__ __forceinline__ void async_store_b128(void* gaddr, uint32_t lds_off) {
    asm volatile("global_store_async_from_lds_b128 %0, %1, off"
                 :: "v"(gaddr), "v"(lds_off) : "memory");
}
__device__ __forceinline__ void wait_async0() {
    asm volatile("s_wait_asynccnt 0" ::: "memory");
}

__global__ __launch_bounds__(RAYS)
void nerf_render_kernel(const float* __restrict__ pos,
                        const float* __restrict__ den,
                        const float* __restrict__ rgb,
                        float* __restrict__ out,
                        int R)
{
    __shared__ __align__(16) uint8_t smem[LDS_TOTAL];
    // Low 32 bits of the flat shared address == LDS byte offset (aperture layout).
    const uint32_t lds_base = (uint32_t)(uintptr_t)&smem[0];

    const int tid  = threadIdx.x;
    const int ray0 = blockIdx.x * RAYS;
    const int ray  = ray0 + tid;

    const char* pos_b = (const char*)(pos + (size_t)ray0 * S_SAMPLES * 3);
    const char* rgb_b = (const char*)(rgb + (size_t)ray0 * S_SAMPLES * 3);
    const char* den_b = (const char*)(den + (size_t)ray0 * S_SAMPLES);
    char*       w_b   = (char*)(out + (size_t)R * 4 + (size_t)ray0 * S_SAMPLES);

    float T = 1.0f, acc = 0.0f, cr = 0.0f, cg = 0.0f, cb = 0.0f;

    for (int k = 0; k < NCHUNK; ++k) {
        const size_t g3 = (size_t)k * 192;   // byte offset into a ray's 3-comp stream
        const size_t g1 = (size_t)k * 64;    // byte offset into a ray's 1-comp stream

        // --- async stage: positions & rgb, 12 x B128 per ray (192 B contiguous) ---
        #pragma unroll
        for (int j = 0; j < 12; ++j) {
            int idx = j * RAYS + tid;
            int r = idx / 12, v = idx % 12;
            async_load_b128(lds_base + LDS_POS + r * POS_STRIDE + v * 16,
                            pos_b + (size_t)r * 576 + g3 + v * 16);
            async_load_b128(lds_base + LDS_RGB + r * RGB_STRIDE + v * 16,
                            rgb_b + (size_t)r * 576 + g3 + v * 16);
        }
        // --- densities, 4 x B128 per ray (64 B contiguous) ---
        #pragma unroll
        for (int j = 0; j < 4; ++j) {
            int idx = j * RAYS + tid;
            int r = idx >> 2, v = idx & 3;
            async_load_b128(lds_base + LDS_DEN + r * DEN_STRIDE + v * 16,
                            den_b + (size_t)r * 192 + g1 + v * 16);
        }
        // --- boundary position (first sample of next chunk), 3 x B32 per ray ---
        if (k < NCHUNK - 1) {
            #pragma unroll
            for (int j = 0; j < 3; ++j) {
                int idx = j * RAYS + tid;
                int r = idx / 3, c = idx % 3;
                async_load_b32(lds_base + LDS_POS + r * POS_STRIDE + 192 + c * 4,
                               pos_b + (size_t)r * 576 + g3 + 192 + c * 4);
            }
        }
        wait_async0();
        __syncthreads();

        // --- per-ray sequential scan out of LDS ---
        const float* lp = (const float*)(smem + LDS_POS + tid * POS_STRIDE);
        const float* lg = (const float*)(smem + LDS_RGB + tid * RGB_STRIDE);
        const float* ld = (const float*)(smem + LDS_DEN + tid * DEN_STRIDE);
        float*       lw = (float*)(smem + LDS_W + tid * W_STRIDE);

        float px = lp[0], py = lp[1], pz = lp[2];
        #pragma unroll
        for (int i = 0; i < CHUNK; ++i) {
            float delta;
            if (k * CHUNK + i == S_SAMPLES - 1) {
                delta = 1.0e10f;                       // FAR_DELTA for last sample
            } else {
                float nx = lp[(i + 1) * 3 + 0];
                float ny = lp[(i + 1) * 3 + 1];
                float nz = lp[(i + 1) * 3 + 2];
                float dx = nx - px, dy = ny - py, dz = nz - pz;
                delta = sqrtf(dx * dx + dy * dy + dz * dz);
                px = nx; py = ny; pz = nz;
            }
            float d = ld[i];
            d = d > 0.0f ? d : 0.0f;                   // relu(densities)
            float alpha = 1.0f - __expf(-d * delta);
            float w = alpha * T;                       // exclusive transmittance
            T = T * (1.0f - alpha + 1.0e-10f);         // cumprod(1 - alpha + EPS)
            acc += w;
            cr += w * lg[i * 3 + 0];
            cg += w * lg[i * 3 + 1];
            cb += w * lg[i * 3 + 2];
            lw[i] = w;
        }
        __syncthreads();

        // --- drain weights chunk: async LDS->global, 4 x B128 per ray ---
        #pragma unroll
        for (int j = 0; j < 4; ++j) {
            int idx = j * RAYS + tid;
            int r = idx >> 2, v = idx & 3;
            async_store_b128(w_b + (size_t)r * 192 + g1 + v * 16,
                             lds_base + LDS_W + r * W_STRIDE + v * 16);
        }
        // Next iteration's s_wait_asynccnt 0 fences these stores before LDS_W reuse.
    }

    // Final small outputs: rendered_rgb [R,3] then accumulation [R]
    out[(size_t)ray * 3 + 0] = cr;
    out[(size_t)ray * 3 + 1] = cg;
    out[(size_t)ray * 3 + 2] = cb;
    out[(size_t)R * 3 + ray] = acc;

    wait_async0();   // ensure last weight drain complete before wave retires
}

extern "C" void kernel_launch(void* const* d_in, const int* in_sizes, int n_in,
                              void* d_out, int out_size, void* d_ws, size_t ws_size,
                              hipStream_t stream) {
    const float* ray_samples = (const float*)d_in[0];   // [R,48,3]
    const float* densities   = (const float*)d_in[1];   // [R,48,1]
    const float* rgb         = (const float*)d_in[2];   // [R,48,3]
    float* out = (float*)d_out;   // [R*3 rendered_rgb | R accumulation | R*48 weights]

    const int R = in_sizes[0] / (S_SAMPLES * 3);        // 262144
    const int grid = R / RAYS;                          // R is a multiple of 64

    nerf_render_kernel<<<dim3(grid), dim3(RAYS), 0, stream>>>(
        ray_samples, densities, rgb, out, R);
}